// RoiPooling_32143535243432
// MI455X (gfx1250) — compile-verified
//
#include <hip/hip_runtime.h>

// ROI-align 7x7 over a (1,128,128,256) fp32 feature map, 4000 ROIs.
// Store-bandwidth-bound: ~200 MB output written once (~8.6 us at 23.3 TB/s),
// 16.8 MB feature map stays L2-resident because the output stream uses
// non-temporal (TH_STORE_NT) 128-bit stores.
//
// Mapping (wave32): block = 224 threads = 7 waves = one (roi, py) row.
//   waveId (0..6) = px, lane (0..31) = 8-channel chunk (lane*8 .. lane*8+7).
// All coordinate math is uniform per wave -> SALU; lanes only do coalesced
// global_load_b128 gathers from L2 and global_store_b128 NT stores.

typedef float f32x4 __attribute__((ext_vector_type(4)));

#define HH 128
#define WW 128
#define CC 256
#define PP 7

__global__ __launch_bounds__(224)
void roi_align7x7_kernel(const float* __restrict__ feat,
                         const float* __restrict__ rois,
                         float* __restrict__ out) {
    const int row  = blockIdx.x;           // (roi, py): 0 .. nroi*7-1
    const int r    = row / PP;
    const int py   = row - r * PP;
    const int lane = threadIdx.x & 31;
    // wave id == px; force into an SGPR so all x-math scalarizes
    const int px   = __builtin_amdgcn_readfirstlane((int)(threadIdx.x >> 5));

    // ROI box (block-uniform -> scalar loads)
    const float ry1 = rois[r * 4 + 0];
    const float rx1 = rois[r * 4 + 1];
    const float ry2 = rois[r * 4 + 2];
    const float rx2 = rois[r * 4 + 3];

    // Sample coordinates (match reference arithmetic order):
    // ys = y1*(H-1) + iy * ((y2-y1)*(H-1)/(ph-1))
    const float dy = (ry2 - ry1) * 127.0f / 6.0f;
    const float dx = (rx2 - rx1) * 127.0f / 6.0f;
    const float ys = ry1 * 127.0f + (float)py * dy;
    const float xs = rx1 * 127.0f + (float)px * dx;

    const float y0f = floorf(ys);
    const float x0f = floorf(xs);
    const float wy  = ys - y0f;
    const float wx  = xs - x0f;

    int iy0 = (int)y0f; iy0 = iy0 < 0 ? 0 : (iy0 > HH - 1 ? HH - 1 : iy0);
    int ix0 = (int)x0f; ix0 = ix0 < 0 ? 0 : (ix0 > WW - 1 ? WW - 1 : ix0);
    const int iy1 = (iy0 + 1 > HH - 1) ? HH - 1 : iy0 + 1;
    const int ix1 = (ix0 + 1 > WW - 1) ? WW - 1 : ix0 + 1;

    const bool valid = (ys >= 0.0f) & (ys <= 127.0f) &
                       (xs >= 0.0f) & (xs <= 127.0f);

    // Per-lane channel chunk: 8 channels = two float4 loads per tap.
    const int c = lane << 3;
    const int o00 = (iy0 * WW + ix0) * CC + c;
    const int o01 = (iy0 * WW + ix1) * CC + c;
    const int o10 = (iy1 * WW + ix0) * CC + c;
    const int o11 = (iy1 * WW + ix1) * CC + c;

    // Gather the 4 bilinear taps (each wave reads 4 contiguous 1KB rows from
    // the L2-resident feature map; default RT temporal hint keeps them hot).
    const f32x4 a00 = *(const f32x4*)(feat + o00);
    const f32x4 b00 = *(const f32x4*)(feat + o00 + 4);
    const f32x4 a01 = *(const f32x4*)(feat + o01);
    const f32x4 b01 = *(const f32x4*)(feat + o01 + 4);
    const f32x4 a10 = *(const f32x4*)(feat + o10);
    const f32x4 b10 = *(const f32x4*)(feat + o10 + 4);
    const f32x4 a11 = *(const f32x4*)(feat + o11);
    const f32x4 b11 = *(const f32x4*)(feat + o11 + 4);

    const float omwx = 1.0f - wx;
    const float omwy = 1.0f - wy;

    f32x4 topA = a00 * omwx + a01 * wx;
    f32x4 botA = a10 * omwx + a11 * wx;
    f32x4 resA = topA * omwy + botA * wy;

    f32x4 topB = b00 * omwx + b01 * wx;
    f32x4 botB = b10 * omwx + b11 * wx;
    f32x4 resB = topB * omwy + botB * wy;

    if (!valid) {
        resA = (f32x4)0.0f;
        resB = (f32x4)0.0f;
    }

    // Output (4000,7,7,256): streamed once -> non-temporal b128 stores so the
    // 200MB output does not evict the 16.8MB feature map from the 192MB L2.
    float* op = out + ((long)row * PP + px) * CC + c;
    __builtin_nontemporal_store(resA, (f32x4*)op);
    __builtin_nontemporal_store(resB, (f32x4*)(op + 4));
}

extern "C" void kernel_launch(void* const* d_in, const int* in_sizes, int n_in,
                              void* d_out, int out_size, void* d_ws, size_t ws_size,
                              hipStream_t stream) {
    (void)n_in; (void)out_size; (void)d_ws; (void)ws_size;
    const float* x    = (const float*)d_in[0];   // (1,128,128,256) f32
    const float* rois = (const float*)d_in[1];   // (4000,4) f32
    float* out        = (float*)d_out;           // (4000,7,7,256) f32

    const int nroi = in_sizes[1] / 4;            // 4000
    const dim3 grid(nroi * PP);                  // one block per (roi, py) row
    roi_align7x7_kernel<<<grid, 224, 0, stream>>>(x, rois, out);
}